// Database_30520037605766
// MI455X (gfx1250) — compile-verified
//
#include <hip/hip_runtime.h>
#include <math.h>

// ---------------------------------------------------------------------------
// Top-2 similarity search, HBM-bandwidth-bound (1.46 GB bank stream, ~63us
// floor at 23.3 TB/s). Uses V_WMMA_F32_16X16X4_F32: M=16 queries, N=16 bank
// columns per tile, K stepped by 4 over D=384 (96 WMMA per tile).
// ---------------------------------------------------------------------------

typedef float v2f __attribute__((ext_vector_type(2)));
typedef float v8f __attribute__((ext_vector_type(8)));

#define QROWS 16
#define WAVES_PER_BLOCK 8
#define MAIN_BLOCKS 256
#define NWAVES (MAIN_BLOCKS * WAVES_PER_BLOCK)

// ---------------------------------------------------------------------------
// Kernel 1: L1-normalize the 16 query rows. One wave per row.
// ---------------------------------------------------------------------------
__global__ __launch_bounds__(512) void l1norm_kernel(const float* __restrict__ q,
                                                     float* __restrict__ qn, int D) {
    const int row = threadIdx.x >> 5;
    const int lane = threadIdx.x & 31;
    if (row >= QROWS) return;
    const float* src = q + (size_t)row * D;
    float s = 0.0f;
    for (int k = lane; k < D; k += 32) s += fabsf(src[k]);
    #pragma unroll
    for (int off = 16; off > 0; off >>= 1) s += __shfl_xor(s, off, 32);
    const float inv = 1.0f / fmaxf(s, 1e-12f);
    float* dst = qn + (size_t)row * D;
    for (int k = lane; k < D; k += 32) dst[k] = src[k] * inv;
}

// ---------------------------------------------------------------------------
// Kernel 2: streaming GEMM + per-wave running top-2 per query row.
// A (16x4 f32, 2 VGPR): lane m=lane&15 holds row m; half-wave selects K pair.
// B (4x16 f32, 2 VGPR): lane n=lane&15 holds col n; half-wave selects K pair.
// C/D (16x16 f32, 8 VGPR): VGPR r holds rows r (lanes 0-15) / r+8 (16-31).
// ---------------------------------------------------------------------------
__global__ __launch_bounds__(256) void simtop_kernel(const float* __restrict__ qn,
                                                     const float* __restrict__ bank,
                                                     const int* __restrict__ startP,
                                                     const int* __restrict__ endP,
                                                     int N, int D,
                                                     float* __restrict__ cand_val,
                                                     int* __restrict__ cand_idx) {
    __shared__ float tile[WAVES_PER_BLOCK][16 * 17];

    const int s = *startP;
    const int e = *endP;
    const int t1 = (s + 15) >> 4;              // tiles covering [0, start)
    const int t2 = (N - e + 15) >> 4;          // tiles covering [end, N)
    const int NT = t1 + t2;

    const int lane = threadIdx.x & 31;
    const int wave = threadIdx.x >> 5;
    const int gwave = blockIdx.x * WAVES_PER_BLOCK + wave;
    const int m = lane & 15;                   // A row / B col for this lane
    const int kh = (lane >> 4) << 1;           // half-wave K offset: 0 or 2
    float* tl = &tile[wave][0];

    float v1 = -INFINITY, v2 = -INFINITY;
    int i1 = 0, i2 = 0;

    const float* arow = qn + (size_t)m * D;

    for (int t = gwave; t < NT; t += NWAVES) {
        const int col0 = (t < t1) ? (t << 4) : (e + ((t - t1) << 4));
        int cc = col0 + m;
        if (cc > N - 1) cc = N - 1;            // clamp (filtered in scan below)

        v8f acc = {};
        #pragma unroll 4
        for (int k0 = 0; k0 < D; k0 += 4) {
            v2f a, b;
            a.x = arow[k0 + kh];
            a.y = arow[k0 + kh + 1];
            const size_t base = (size_t)(k0 + kh) * (size_t)N + (size_t)cc;
            b.x = bank[base];
            b.y = bank[base + (size_t)N];
            acc = __builtin_amdgcn_wmma_f32_16x16x4_f32(
                /*neg_a=*/false, a, /*neg_b=*/false, b,
                /*c_mod=*/(short)0, acc, /*reuse_a=*/false, /*reuse_b=*/false);
        }

        // Spill 16x16 sim tile to per-wave LDS (stride 17 -> bank-conflict-free)
        const int hi = lane >> 4;
        #pragma unroll
        for (int r = 0; r < 8; ++r)
            tl[(r + hi * 8) * 17 + m] = acc[r];
        asm volatile("s_wait_dscnt 0x0" ::: "memory");   // intra-wave DS RAW
        __builtin_amdgcn_wave_barrier();

        if (lane < 16) {                       // lane = query row
            #pragma unroll
            for (int j = 0; j < 16; ++j) {
                const float v = tl[lane * 17 + j];
                const int c = col0 + j;
                const bool valid = (c < s) || (c >= e && c < N);
                if (valid) {
                    if (v > v1)      { v2 = v1; i2 = i1; v1 = v; i1 = c; }
                    else if (v > v2) { v2 = v;  i2 = c; }
                }
            }
        }
        __builtin_amdgcn_wave_barrier();
    }

    if (lane < 16) {
        const int o = (gwave * QROWS + lane) * 2;
        cand_val[o]     = v1;
        cand_val[o + 1] = v2;
        cand_idx[o]     = i1;
        cand_idx[o + 1] = i2;
    }
}

// ---------------------------------------------------------------------------
// Kernel 3: merge per-wave candidates into final top-2 per row.
// blockIdx.x = query row. Output: values [16*2] then indices-as-float [16*2].
// ---------------------------------------------------------------------------
__global__ __launch_bounds__(256) void reduce_kernel(const float* __restrict__ cand_val,
                                                     const int* __restrict__ cand_idx,
                                                     float* __restrict__ out,
                                                     int nwaves) {
    __shared__ float sv1[256], sv2[256];
    __shared__ int   si1[256], si2[256];

    const int r = blockIdx.x;
    const int tid = threadIdx.x;

    float v1 = -INFINITY, v2 = -INFINITY;
    int i1 = 0, i2 = 0;

    for (int w = tid; w < nwaves; w += 256) {
        const int o = (w * QROWS + r) * 2;
        #pragma unroll
        for (int k = 0; k < 2; ++k) {
            const float v = cand_val[o + k];
            const int   c = cand_idx[o + k];
            if (v > v1)      { v2 = v1; i2 = i1; v1 = v; i1 = c; }
            else if (v > v2) { v2 = v;  i2 = c; }
        }
    }
    sv1[tid] = v1; sv2[tid] = v2; si1[tid] = i1; si2[tid] = i2;
    __syncthreads();

    for (int st = 128; st > 0; st >>= 1) {
        if (tid < st) {
            float a1 = sv1[tid + st], a2 = sv2[tid + st];
            int   b1 = si1[tid + st], b2 = si2[tid + st];
            float w1 = sv1[tid], w2 = sv2[tid];
            int   j1 = si1[tid], j2 = si2[tid];
            if (a1 > w1)      { w2 = w1; j2 = j1; w1 = a1; j1 = b1; }
            else if (a1 > w2) { w2 = a1; j2 = b1; }
            if (a2 > w2)      {
                if (a2 > w1)  { w2 = w1; j2 = j1; w1 = a2; j1 = b2; }
                else          { w2 = a2; j2 = b2; }
            }
            sv1[tid] = w1; sv2[tid] = w2; si1[tid] = j1; si2[tid] = j2;
        }
        __syncthreads();
    }

    if (tid == 0) {
        out[r * 2 + 0] = sv1[0];
        out[r * 2 + 1] = sv2[0];
        out[2 * QROWS + r * 2 + 0] = (float)si1[0];
        out[2 * QROWS + r * 2 + 1] = (float)si2[0];
    }
}

// ---------------------------------------------------------------------------
extern "C" void kernel_launch(void* const* d_in, const int* in_sizes, int n_in,
                              void* d_out, int out_size, void* d_ws, size_t ws_size,
                              hipStream_t stream) {
    const float* q      = (const float*)d_in[0];   // [16, 384]
    const float* bank   = (const float*)d_in[1];   // [384, N]
    const int*   startP = (const int*)d_in[2];
    const int*   endP   = (const int*)d_in[3];

    const int D = in_sizes[0] / QROWS;             // 384
    const int N = in_sizes[1] / D;                 // 1,000,000

    float* qn       = (float*)d_ws;                         // 16*D floats
    float* cand_val = qn + (size_t)QROWS * D;               // NWAVES*16*2 floats
    int*   cand_idx = (int*)(cand_val + (size_t)NWAVES * QROWS * 2);

    l1norm_kernel<<<1, 512, 0, stream>>>(q, qn, D);
    simtop_kernel<<<MAIN_BLOCKS, WAVES_PER_BLOCK * 32, 0, stream>>>(
        qn, bank, startP, endP, N, D, cand_val, cand_idx);
    reduce_kernel<<<QROWS, 256, 0, stream>>>(cand_val, cand_idx, (float*)d_out, NWAVES);
}